// DiceCoefficient_73821897884105
// MI455X (gfx1250) — compile-verified
//
#include <hip/hip_runtime.h>
#include <hip/hip_bf16.h>

#define NUM_GROUPS 64
#define EPS 1e-5f

typedef __attribute__((ext_vector_type(2))) float v2f;
typedef __attribute__((ext_vector_type(8))) float v8f;

// Wave32 all-lane sum via V_WMMA_F32_16X16X4_F32.
// A (16x4 f32): lanes 0-15 hold K=0 (vgpr0) / K=1 (vgpr1); lanes 16-31 hold K=2/K=3.
// Put partial in vgpr0, zero in vgpr1, B = ones  =>  rowsum[m] = p[m] + p[m+16].
// D (16x16 f32): vgpr r = rowsum[r] on lanes 0-15, rowsum[r+8] on lanes 16-31.
// Sum the 8 D vgprs per lane, then shfl_xor(16) to combine halves.
// Requires EXEC all ones: call with every thread of the block active.
__device__ __forceinline__ float wave_reduce_sum(float p) {
    v2f a; a[0] = p;    a[1] = 0.0f;
    v2f b; b[0] = 1.0f; b[1] = 1.0f;
    v8f c = {};
    v8f d = __builtin_amdgcn_wmma_f32_16x16x4_f32(
        /*neg_a=*/false, a, /*neg_b=*/false, b,
        /*c_mod=*/(short)0, c, /*reuse_a=*/false, /*reuse_b=*/false);
    float s = d[0] + d[1] + d[2] + d[3] + d[4] + d[5] + d[6] + d[7];
    s += __shfl_xor(s, 16, 32);
    return s;
}

// Reduce three per-thread partials across the whole block (result valid in thread 0).
__device__ __forceinline__ void block_reduce3(float& a, float& b, float& c, float* lds) {
    a = wave_reduce_sum(a);
    b = wave_reduce_sum(b);
    c = wave_reduce_sum(c);
    const int lane = threadIdx.x & 31;
    const int wave = threadIdx.x >> 5;
    if (lane == 0) {
        lds[wave * 3 + 0] = a;
        lds[wave * 3 + 1] = b;
        lds[wave * 3 + 2] = c;
    }
    __syncthreads();
    if (threadIdx.x == 0) {
        const int nw = blockDim.x >> 5;
        float sa = 0.f, sb = 0.f, sc = 0.f;
        for (int w = 0; w < nw; ++w) {
            sa += lds[w * 3 + 0];
            sb += lds[w * 3 + 1];
            sc += lds[w * 3 + 2];
        }
        a = sa; b = sb; c = sc;
    }
}

// Kernel 1: one block per teacher instance. Streams preds_T/gt_T rows (float4),
// computes dice loss and caches sum(x*x) for reuse in the student pass.
__global__ void teacher_dice_kernel(const float* __restrict__ pT,
                                    const float* __restrict__ gT,
                                    float* __restrict__ loss_T,
                                    float* __restrict__ xx_T,
                                    int hw) {
    const int row = blockIdx.x;
    const int hw4 = hw >> 2;
    const float4* __restrict__ x4 = (const float4*)(pT + (size_t)row * hw);
    const float4* __restrict__ t4 = (const float4*)(gT + (size_t)row * hw);

    float inter = 0.f, xx = 0.f, tt = 0.f;
    for (int i = threadIdx.x; i < hw4; i += blockDim.x) {
        const float4 x = x4[i];
        const float4 t = t4[i];
        inter = fmaf(x.x, t.x, fmaf(x.y, t.y, fmaf(x.z, t.z, fmaf(x.w, t.w, inter))));
        xx    = fmaf(x.x, x.x, fmaf(x.y, x.y, fmaf(x.z, x.z, fmaf(x.w, x.w, xx))));
        tt    = fmaf(t.x, t.x, fmaf(t.y, t.y, fmaf(t.z, t.z, fmaf(t.w, t.w, tt))));
    }

    __shared__ float lds[32 * 3];
    block_reduce3(inter, xx, tt, lds);
    if (threadIdx.x == 0) {
        loss_T[row] = 1.0f - 2.0f * inter / (xx + tt + EPS);
        xx_T[row]   = xx;
    }
}

// Kernel 2: per-group argmin over teacher losses (first-min semantics like jnp.argmin).
__global__ void group_argmin_kernel(const float* __restrict__ loss_T,
                                    const int* __restrict__ gt_inds_T,
                                    int* __restrict__ best_idx,
                                    int* __restrict__ present,
                                    int nT) {
    const int g = threadIdx.x;
    if (g >= NUM_GROUPS) return;
    float best = __builtin_huge_valf();
    int   bidx = 0;
    int   found = 0;
    for (int i = 0; i < nT; ++i) {
        if (gt_inds_T[i] == g) {
            found = 1;
            const float l = loss_T[i];
            if (l < best) { best = l; bidx = i; }
        }
    }
    best_idx[g] = bidx;
    present[g]  = found;
}

// Kernel 3: one block per student instance; dice against the matched (nms'd) teacher row.
// sum(t*t) of the teacher row is reused from kernel 1.
__global__ void student_dice_kernel(const float* __restrict__ pT,
                                    const float* __restrict__ pS,
                                    const int* __restrict__ gt_inds_S,
                                    const int* __restrict__ best_idx,
                                    const int* __restrict__ present,
                                    const float* __restrict__ xx_T,
                                    float* __restrict__ per_pair,
                                    int hw) {
    const int s = blockIdx.x;
    const int g = gt_inds_S[s];           // uniform across block
    if (!present[g]) {                    // uniform branch: WMMA never sees partial EXEC
        if (threadIdx.x == 0) per_pair[s] = 0.0f;
        return;
    }
    const int trow = best_idx[g];
    const int hw4  = hw >> 2;
    const float4* __restrict__ x4 = (const float4*)(pS + (size_t)s    * hw);
    const float4* __restrict__ t4 = (const float4*)(pT + (size_t)trow * hw);

    float inter = 0.f, ss = 0.f, zero = 0.f;
    for (int i = threadIdx.x; i < hw4; i += blockDim.x) {
        const float4 x = x4[i];
        const float4 t = t4[i];
        inter = fmaf(x.x, t.x, fmaf(x.y, t.y, fmaf(x.z, t.z, fmaf(x.w, t.w, inter))));
        ss    = fmaf(x.x, x.x, fmaf(x.y, x.y, fmaf(x.z, x.z, fmaf(x.w, x.w, ss))));
    }

    __shared__ float lds[32 * 3];
    block_reduce3(inter, ss, zero, lds);
    if (threadIdx.x == 0) {
        per_pair[s] = 1.0f - 2.0f * inter / (ss + xx_T[trow] + EPS);
    }
}

// Kernel 4: deterministic fixed-order tree sum of the per-pair losses -> scalar output.
__global__ void final_sum_kernel(const float* __restrict__ per_pair,
                                 float* __restrict__ out, int n) {
    __shared__ float sdata[256];
    const int tid = threadIdx.x;
    sdata[tid] = (tid < n) ? per_pair[tid] : 0.0f;
    __syncthreads();
    for (int s = blockDim.x >> 1; s > 0; s >>= 1) {
        if (tid < s) sdata[tid] += sdata[tid + s];
        __syncthreads();
    }
    if (tid == 0) out[0] = sdata[0];   // LOSS_WEIGHT == 1.0
}

extern "C" void kernel_launch(void* const* d_in, const int* in_sizes, int n_in,
                              void* d_out, int out_size, void* d_ws, size_t ws_size,
                              hipStream_t stream) {
    const float* preds_T   = (const float*)d_in[0];
    const float* preds_S   = (const float*)d_in[1];
    // d_in[2] = im_ind (unused), d_in[4] = gt_S (unused), d_in[5] = iter (unused)
    const float* gt_T      = (const float*)d_in[3];
    const int*   gt_inds_T = (const int*)d_in[6];
    const int*   gt_inds_S = (const int*)d_in[7];
    float*       out       = (float*)d_out;

    const int nT = in_sizes[6];           // 256
    const int nS = in_sizes[7];           // 128
    const int hw = in_sizes[0] / nT;      // 65536 (divisible by 4)

    // Workspace layout
    float* loss_T   = (float*)d_ws;
    float* xx_T     = loss_T + nT;
    int*   best_idx = (int*)(xx_T + nT);
    int*   present  = best_idx + NUM_GROUPS;
    float* per_pair = (float*)(present + NUM_GROUPS);

    teacher_dice_kernel<<<nT, 256, 0, stream>>>(preds_T, gt_T, loss_T, xx_T, hw);
    group_argmin_kernel<<<1, NUM_GROUPS, 0, stream>>>(loss_T, gt_inds_T, best_idx, present, nT);
    student_dice_kernel<<<nS, 256, 0, stream>>>(preds_T, preds_S, gt_inds_S, best_idx,
                                                present, xx_T, per_pair, hw);
    final_sum_kernel<<<1, 256, 0, stream>>>(per_pair, out, nS);
}